// Sentence_Embedding_4183298146588
// MI455X (gfx1250) — compile-verified
//
#include <hip/hip_runtime.h>
#include <hip/hip_bf16.h>

typedef __attribute__((ext_vector_type(2))) float v2f;
typedef __attribute__((ext_vector_type(4))) float v4f;
typedef __attribute__((ext_vector_type(8))) float v8f;

#define BATCH  2048
#define MAXLEN 200
#define HIDDEN 256
#define H4     (HIDDEN / 4)   // 64 float4 groups per row

// ---------------------------------------------------------------------------
// Kernel 1: pooled[b][h] = (1/L) * sum_l emb[x[b][l]][h]
// One 64-thread block (2 waves) per batch row; each thread owns one float4
// column group -> global_load_b128, 1KB fully-coalesced burst per token row.
// Token index is wave-uniform -> scalar load. Table (102MB) is L2-resident
// (192MB L2), so this kernel is L2-BW bound; b128 loads minimize issue count.
// ---------------------------------------------------------------------------
__global__ void __launch_bounds__(H4)
pool_kernel(const int* __restrict__ x,
            const float* __restrict__ emb,
            float* __restrict__ pooled) {
    const int b  = blockIdx.x;
    const int h4 = threadIdx.x;                       // 0..63
    const int* xrow = x + b * MAXLEN;
    const v4f* __restrict__ emb4 = (const v4f*)emb;   // [VOCAB][H4]

    v4f acc = {};
#pragma unroll 4
    for (int l = 0; l < MAXLEN; ++l) {
        const int idx = xrow[l];                      // uniform -> s_load
        acc += emb4[(size_t)idx * H4 + h4];           // global_load_b128
    }
    acc *= (1.0f / MAXLEN);
    ((v4f*)pooled)[(size_t)b * H4 + h4] = acc;        // global_store_b128
}

// ---------------------------------------------------------------------------
// Kernel 2: out[m][n] = sum_k pooled[m][k] * W[n][k] + bias[n]
// One wave32 per 16x16 output tile; K-loop of V_WMMA_F32_16X16X4_F32.
//
// ISA VGPR layouts (CDNA5 ISA):
//   A 16x4  : lanes 0-15 -> M=lane,    v0=K0, v1=K1
//             lanes16-31 -> M=lane-16, v0=K2, v1=K3
//   B 4x16  : lanes 0-15 -> N=lane,    v0=K0, v1=K1
//             lanes16-31 -> N=lane-16, v0=K2, v1=K3
//   C/D 16x16: VGPR r -> M = r + 8*(lane>=16), N = lane&15
// ---------------------------------------------------------------------------
__global__ void __launch_bounds__(32)
gemm_kernel(const float* __restrict__ s,
            const float* __restrict__ W,
            const float* __restrict__ bias,
            float* __restrict__ out) {
    const int lane  = threadIdx.x;        // 0..31, all lanes active (EXEC=~0)
    const int half  = lane >> 4;          // 0: K pair {0,1}; 1: K pair {2,3}
    const int l15   = lane & 15;
    const int mTile = blockIdx.x;         // BATCH/16 tiles
    const int nTile = blockIdx.y;         // HIDDEN/16 tiles

    const int mRow = mTile * 16 + l15;    // A-matrix row held by this lane
    const int nCol = nTile * 16 + l15;    // B-matrix col held by this lane

    const float* __restrict__ srow = s + (size_t)mRow * HIDDEN;
    const float* __restrict__ wrow = W + (size_t)nCol * HIDDEN;

    v8f acc = {};
#pragma unroll 8
    for (int k = 0; k < HIDDEN; k += 4) {
        const int kk = k + 2 * half;
        v2f a;                            // A[M=mRow][kk], A[M=mRow][kk+1]
        a.x = srow[kk];
        a.y = srow[kk + 1];
        v2f bmat;                         // B[kk][nCol] = W[nCol][kk]
        bmat.x = wrow[kk];
        bmat.y = wrow[kk + 1];
        // (neg_a, A, neg_b, B, c_mod, C, reuse_a, reuse_b)
        acc = __builtin_amdgcn_wmma_f32_16x16x4_f32(
            false, a, false, bmat, (short)0, acc, false, false);
    }

    const float bv = bias[nCol];
#pragma unroll
    for (int r = 0; r < 8; ++r) {
        const int m = mTile * 16 + r + half * 8;
        out[(size_t)m * HIDDEN + nCol] = acc[r] + bv;
    }
}

// ---------------------------------------------------------------------------
extern "C" void kernel_launch(void* const* d_in, const int* in_sizes, int n_in,
                              void* d_out, int out_size, void* d_ws, size_t ws_size,
                              hipStream_t stream) {
    const int*   x    = (const int*)d_in[0];     // [BATCH, MAXLEN] int32
    const float* emb  = (const float*)d_in[1];   // [VOCAB, HIDDEN] f32
    const float* W    = (const float*)d_in[2];   // [HIDDEN, HIDDEN] f32
    const float* bias = (const float*)d_in[3];   // [HIDDEN] f32
    float*       out  = (float*)d_out;           // [BATCH, HIDDEN] f32
    float*       pooled = (float*)d_ws;          // [BATCH, HIDDEN] f32 (2 MB)

    pool_kernel<<<BATCH, H4, 0, stream>>>(x, emb, pooled);

    dim3 grid(BATCH / 16, HIDDEN / 16);
    gemm_kernel<<<grid, 32, 0, stream>>>(pooled, W, bias, out);
}